// GCNAE_2207613190405
// MI455X (gfx1250) — compile-verified
//
#include <hip/hip_runtime.h>
#include <hip/hip_bf16.h>

#define N_NODES 12288
#define N_EDGES 393216
#define IN_FEAT 2000
#define HID 64

typedef __attribute__((ext_vector_type(2))) float v2f;
typedef __attribute__((ext_vector_type(8))) float v8f;

// -------- CDNA5 async global->LDS path (guarded; falls back to sync staging) --------
#if defined(__HIP_DEVICE_COMPILE__) && __has_builtin(__builtin_amdgcn_global_load_async_to_lds_b32)
#define HAVE_ASYNC_LDS 1
#else
#define HAVE_ASYNC_LDS 0
#endif

#if HAVE_ASYNC_LDS
#if __has_builtin(__builtin_amdgcn_s_wait_asynccnt)
#define WAIT_ASYNCCNT(n) __builtin_amdgcn_s_wait_asynccnt(n)
#else
#define WAIT_ASYNCCNT(n) asm volatile("s_wait_asynccnt %0" ::"i"(n) : "memory")
#endif

typedef __attribute__((address_space(1))) int gas_int;  // global
typedef __attribute__((address_space(3))) int las_int;  // LDS

__device__ __forceinline__ void async_ld_b32(const float* g, float* lds_generic) {
    // global (as1) src, LDS (as3) dst, imm offset, cpol
    __builtin_amdgcn_global_load_async_to_lds_b32(
        (gas_int*)(unsigned long long)(uintptr_t)g,   // flat addr of global == as1 addr
        (las_int*)lds_generic,                        // addrspacecast flat -> group
        0, 0);
}
#endif

// ---------------------------------------------------------------- utilities

__global__ void zero_kernel(float* __restrict__ p, int n) {
    int i = blockIdx.x * blockDim.x + threadIdx.x;
    if (i < n) p[i] = 0.0f;
}

__global__ void degree_kernel(const int* __restrict__ src, const int* __restrict__ dst,
                              float* __restrict__ deg_out, float* __restrict__ deg_in) {
    int e = blockIdx.x * blockDim.x + threadIdx.x;
    if (e < N_EDGES) {
        unsafeAtomicAdd(&deg_out[src[e]], 1.0f);
        unsafeAtomicAdd(&deg_in[dst[e]], 1.0f);
    }
}

__global__ void norm_kernel(float* __restrict__ deg_out, float* __restrict__ deg_in) {
    int i = blockIdx.x * blockDim.x + threadIdx.x;
    if (i < N_NODES) {
        deg_out[i] = rsqrtf(fmaxf(deg_out[i], 1.0f));
        deg_in[i]  = rsqrtf(fmaxf(deg_in[i], 1.0f));
    }
}

// m = (h_raw[src] * out_norm[src]) * ew ; agg[dst] += m   (thread per (edge,feature))
// out_norm folded here so the GEMM can async-copy raw rows.
__global__ void scatter_kernel(const float* __restrict__ h, const float* __restrict__ out_norm,
                               const float* __restrict__ ew,
                               const int* __restrict__ src, const int* __restrict__ dst,
                               float* __restrict__ agg) {
    long long gid = (long long)blockIdx.x * blockDim.x + threadIdx.x;
    int e = (int)(gid >> 6);
    int f = (int)(gid & 63);
    if (e < N_EDGES) {
        int s = src[e], d = dst[e];
        float v = h[(size_t)s * HID + f] * (out_norm[s] * ew[e]);
        unsafeAtomicAdd(&agg[(size_t)d * HID + f], v);
    }
}

// x = agg * in_norm + b
__global__ void finalize_kernel(const float* __restrict__ agg, const float* __restrict__ in_norm,
                                const float* __restrict__ b, float* __restrict__ x) {
    int gid = blockIdx.x * blockDim.x + threadIdx.x;
    if (gid < N_NODES * HID) {
        int i = gid >> 6, f = gid & 63;
        x[gid] = agg[gid] * in_norm[i] + b[f];
    }
}

// ------------------------------------------------- WMMA GEMM: X @ W -> H[M,64]  (raw, unscaled)
// Block: 256 threads (8 waves), 64 rows/block, 4x4 grid of 16x16 tiles, 2 tiles/wave.
// K staged in 16-wide chunks. A LDS [row][k] stride 18, W LDS transposed [col][k] stride 18.
// Async path: double-buffered LDS, GLOBAL_LOAD_ASYNC_TO_LDS_B32 overlaps chunk c+1 with chunk c.
__global__ void wmma_gemm64(const float* __restrict__ X, const float* __restrict__ W,
                            float* __restrict__ H, int K) {
#if HAVE_ASYNC_LDS
    __shared__ float As[2][64 * 18];
    __shared__ float Bs[2][64 * 18];
#else
    __shared__ float As[1][64 * 18];
    __shared__ float Bs[1][64 * 18];
#endif

    const int tid  = threadIdx.x;
    const int lane = tid & 31;
    const int w    = tid >> 5;
    const int half = lane >> 4;
    const int r    = lane & 15;
    const int mt   = w & 3;
    const int nt0  = (w >> 2) << 1;        // 0 or 2
    const int rowBase = blockIdx.x * 64;

    v8f acc0 = {}, acc1 = {};

    // 4 micro-steps of K=4 over one staged 16-wide chunk
    auto compute16 = [&](const float* as, const float* bs) {
        #pragma unroll
        for (int kk = 0; kk < 16; kk += 4) {
            int kb = kk + 2 * half;        // K = vgpr + 2*lane_half
            v2f a  = *(const v2f*)&as[(mt * 16 + r) * 18 + kb];
            v2f b0 = *(const v2f*)&bs[((nt0    ) * 16 + r) * 18 + kb];
            v2f b1 = *(const v2f*)&bs[((nt0 + 1) * 16 + r) * 18 + kb];
            acc0 = __builtin_amdgcn_wmma_f32_16x16x4_f32(false, a, false, b0, (short)0, acc0, false, false);
            acc1 = __builtin_amdgcn_wmma_f32_16x16x4_f32(false, a, false, b1, (short)0, acc1, false, false);
        }
    };

#if HAVE_ASYNC_LDS
    // Each thread issues exactly 8 async b32 copies per chunk (4 for A, 4 for W^T).
    auto issue = [&](int k0, int buf) {
        #pragma unroll
        for (int i = 0; i < 4; ++i) {
            int idx = tid + i * 256;
            int row = idx >> 4, kk = idx & 15;
            async_ld_b32(&X[(size_t)(rowBase + row) * K + k0 + kk], &As[buf][row * 18 + kk]);
        }
        #pragma unroll
        for (int i = 0; i < 4; ++i) {
            int idx = tid + i * 256;
            int kk = idx >> 6, c = idx & 63;
            async_ld_b32(&W[(size_t)(k0 + kk) * HID + c], &Bs[buf][c * 18 + kk]);
        }
    };

    const int nch = K >> 4;
    issue(0, 0);
    for (int c = 0; c < nch; ++c) {
        const int cur = c & 1;
        if (c + 1 < nch) {
            issue((c + 1) << 4, cur ^ 1);  // prefetch next chunk into other buffer
            WAIT_ASYNCCNT(8);              // current chunk's 8 copies done (in-order)
        } else {
            WAIT_ASYNCCNT(0);
        }
        __syncthreads();                   // chunk c visible to all waves
        compute16(As[cur], Bs[cur]);
        __syncthreads();                   // all reads of this buffer done before reuse
    }
#else
    for (int k0 = 0; k0 < K; k0 += 16) {
        __syncthreads();
        #pragma unroll
        for (int i = 0; i < 4; ++i) {
            int idx = tid + i * 256;
            int row = idx >> 4, kk = idx & 15;
            As[0][row * 18 + kk] = X[(size_t)(rowBase + row) * K + k0 + kk];
        }
        #pragma unroll
        for (int i = 0; i < 4; ++i) {
            int idx = tid + i * 256;
            int kk = idx >> 6, c = idx & 63;
            Bs[0][c * 18 + kk] = W[(size_t)(k0 + kk) * HID + c];
        }
        __syncthreads();
        compute16(As[0], Bs[0]);
    }
#endif

    // store C: VGPR v -> M = v + 8*half, N = r
    #pragma unroll
    for (int v = 0; v < 8; ++v) {
        int row  = rowBase + mt * 16 + v + 8 * half;
        int col0 = nt0 * 16 + r;
        H[(size_t)row * HID + col0]      = acc0[v];
        H[(size_t)row * HID + col0 + 16] = acc1[v];
    }
}

// ------------------------------------------------- adj = x @ x^T  (64x64 tile / block)
__global__ void wmma_adj(const float* __restrict__ Xv, float* __restrict__ adj) {
    __shared__ float La[64 * 68];
    __shared__ float Lb[64 * 68];

    const int tid  = threadIdx.x;
    const int lane = tid & 31;
    const int w    = tid >> 5;
    const int half = lane >> 4;
    const int r    = lane & 15;
    const int mt   = w & 3;
    const int nt0  = (w >> 2) << 1;
    const int i0 = blockIdx.y * 64;
    const int j0 = blockIdx.x * 64;

    // stage 64x64 row blocks of x for both operands (row-major = both A and B^T layout)
    #pragma unroll
    for (int i = 0; i < 4; ++i) {
        int idx = tid + i * 256;           // 1024 float4 slots
        int row = idx >> 4;
        int c4  = (idx & 15) << 2;
        float4 va = *(const float4*)&Xv[(size_t)(i0 + row) * HID + c4];
        *(float4*)&La[row * 68 + c4] = va;
        float4 vb = *(const float4*)&Xv[(size_t)(j0 + row) * HID + c4];
        *(float4*)&Lb[row * 68 + c4] = vb;
    }
    __syncthreads();

    v8f acc0 = {}, acc1 = {};
    #pragma unroll
    for (int kk = 0; kk < HID; kk += 4) {
        int kb = kk + 2 * half;
        v2f a  = *(const v2f*)&La[(mt * 16 + r) * 68 + kb];
        v2f b0 = *(const v2f*)&Lb[((nt0    ) * 16 + r) * 68 + kb];
        v2f b1 = *(const v2f*)&Lb[((nt0 + 1) * 16 + r) * 68 + kb];
        acc0 = __builtin_amdgcn_wmma_f32_16x16x4_f32(false, a, false, b0, (short)0, acc0, false, false);
        acc1 = __builtin_amdgcn_wmma_f32_16x16x4_f32(false, a, false, b1, (short)0, acc1, false, false);
    }

    // 604 MB output stream: bypass caches with non-temporal stores
    #pragma unroll
    for (int v = 0; v < 8; ++v) {
        size_t row = (size_t)(i0 + mt * 16 + v + 8 * half);
        size_t col = (size_t)(j0 + nt0 * 16 + r);
        __builtin_nontemporal_store(acc0[v], &adj[row * N_NODES + col]);
        __builtin_nontemporal_store(acc1[v], &adj[row * N_NODES + col + 16]);
    }
}

// ---------------------------------------------------------------- launcher

extern "C" void kernel_launch(void* const* d_in, const int* in_sizes, int n_in,
                              void* d_out, int out_size, void* d_ws, size_t ws_size,
                              hipStream_t stream) {
    const float* features = (const float*)d_in[0];
    const float* ew       = (const float*)d_in[1];
    const float* W1       = (const float*)d_in[2];
    const float* b1       = (const float*)d_in[3];
    const float* W2       = (const float*)d_in[4];
    const float* b2       = (const float*)d_in[5];
    const int*   src      = (const int*)d_in[6];
    const int*   dst      = (const int*)d_in[7];

    float* out = (float*)d_out;
    float* ws  = (float*)d_ws;

    float* deg_out = ws;                         // N  (becomes out_norm)
    float* deg_in  = ws + N_NODES;               // N  (becomes in_norm)
    float* h       = ws + 2 * N_NODES;           // 64N (raw X@W)
    float* agg     = h + (size_t)HID * N_NODES;  // 64N
    float* x1      = agg + (size_t)HID * N_NODES;// 64N

    float* adj = out;
    float* x2  = out + (size_t)N_NODES * N_NODES;

    const int T = 256;

    // degrees + norms
    zero_kernel<<<(2 * N_NODES + T - 1) / T, T, 0, stream>>>(deg_out, 2 * N_NODES);
    degree_kernel<<<(N_EDGES + T - 1) / T, T, 0, stream>>>(src, dst, deg_out, deg_in);
    norm_kernel<<<(N_NODES + T - 1) / T, T, 0, stream>>>(deg_out, deg_in);

    // ---- layer 1 (h = features @ W1, out_norm applied in scatter)
    wmma_gemm64<<<N_NODES / 64, T, 0, stream>>>(features, W1, h, IN_FEAT);
    zero_kernel<<<(N_NODES * HID + T - 1) / T, T, 0, stream>>>(agg, N_NODES * HID);
    scatter_kernel<<<(int)(((long long)N_EDGES * HID) / T), T, 0, stream>>>(h, deg_out, ew, src, dst, agg);
    finalize_kernel<<<(N_NODES * HID + T - 1) / T, T, 0, stream>>>(agg, deg_in, b1, x1);

    // ---- layer 2
    wmma_gemm64<<<N_NODES / 64, T, 0, stream>>>(x1, W2, h, HID);
    zero_kernel<<<(N_NODES * HID + T - 1) / T, T, 0, stream>>>(agg, N_NODES * HID);
    scatter_kernel<<<(int)(((long long)N_EDGES * HID) / T), T, 0, stream>>>(h, deg_out, ew, src, dst, agg);
    finalize_kernel<<<(N_NODES * HID + T - 1) / T, T, 0, stream>>>(agg, deg_in, b2, x2);

    // ---- decoder: adj = x2 @ x2^T
    dim3 grid(N_NODES / 64, N_NODES / 64);
    wmma_adj<<<grid, T, 0, stream>>>(x2, adj);
}